// VisionPatchEmbedder_54065048322461
// MI455X (gfx1250) — compile-verified
//
#include <hip/hip_runtime.h>

// Problem dims (fixed by the reference)
#define B_DIM 8
#define L_DIM 4096
#define PD_DIM 588            // K of the GEMM
#define D_DIM 1152            // N of the GEMM
#define M_TOT (B_DIM * L_DIM) // 32768
#define K_TOT PD_DIM
#define N_TOT D_DIM
#define NK_STEPS (K_TOT / 4)  // 147 = 3 * 49

typedef __attribute__((ext_vector_type(2))) float v2f;
typedef __attribute__((ext_vector_type(8))) float v8f;

struct Frag {
    v2f a0, a1;           // two 16x4 A fragments (raw pixel values)
    v2f b0, b1, b2, b3;   // four 4x16 B fragments (W values)
};

__device__ __forceinline__ Frag load_frag(const float* __restrict__ aRow0,
                                          const float* __restrict__ aRow1,
                                          const float* __restrict__ bRow0,
                                          const float* __restrict__ bRow1,
                                          const float* __restrict__ bRow2,
                                          const float* __restrict__ bRow3,
                                          int k) {
    Frag f;
    f.a0 = *(const v2f*)(aRow0 + k);
    f.a1 = *(const v2f*)(aRow1 + k);
    f.b0 = *(const v2f*)(bRow0 + k);
    f.b1 = *(const v2f*)(bRow1 + k);
    f.b2 = *(const v2f*)(bRow2 + k);
    f.b3 = *(const v2f*)(bRow3 + k);
    return f;
}

__device__ __forceinline__ void do_mma(const Frag& f, v8f acc[2][4]) {
    // normalize [0,1] -> [-1,1] at consume time (one v_pk_fma_f32 each)
    v2f a0 = f.a0 * 2.0f - 1.0f;
    v2f a1 = f.a1 * 2.0f - 1.0f;
    // 8 args: (neg_a, A, neg_b, B, c_mod, C, reuse_a, reuse_b)
    acc[0][0] = __builtin_amdgcn_wmma_f32_16x16x4_f32(false, a0, false, f.b0, (short)0, acc[0][0], false, false);
    acc[0][1] = __builtin_amdgcn_wmma_f32_16x16x4_f32(false, a0, false, f.b1, (short)0, acc[0][1], false, false);
    acc[0][2] = __builtin_amdgcn_wmma_f32_16x16x4_f32(false, a0, false, f.b2, (short)0, acc[0][2], false, false);
    acc[0][3] = __builtin_amdgcn_wmma_f32_16x16x4_f32(false, a0, false, f.b3, (short)0, acc[0][3], false, false);
    acc[1][0] = __builtin_amdgcn_wmma_f32_16x16x4_f32(false, a1, false, f.b0, (short)0, acc[1][0], false, false);
    acc[1][1] = __builtin_amdgcn_wmma_f32_16x16x4_f32(false, a1, false, f.b1, (short)0, acc[1][1], false, false);
    acc[1][2] = __builtin_amdgcn_wmma_f32_16x16x4_f32(false, a1, false, f.b2, (short)0, acc[1][2], false, false);
    acc[1][3] = __builtin_amdgcn_wmma_f32_16x16x4_f32(false, a1, false, f.b3, (short)0, acc[1][3], false, false);
}

// out[m,n] = sum_k (2*pix[m,k]-1) * W[n,k]  +  posemb(m,n)
// Block: 256 threads (8 waves). Block tile: 128(M) x 128(N).
// Wave grid in block: 4 (M) x 2 (N); each wave: 32(M) x 64(N) = 2x4 WMMA tiles.
__global__ __launch_bounds__(256)
void patch_embed_wmma_f32(const float* __restrict__ pix,   // [M, K]
                          const int*   __restrict__ ids,   // [M, 2]
                          const unsigned char* __restrict__ mask, // [M]
                          const float* __restrict__ W,     // [N, K]
                          const float* __restrict__ ptab,  // [2, 64, N]
                          float* __restrict__ out)         // [M, N]
{
    const int lane  = threadIdx.x & 31;
    const int wave  = threadIdx.x >> 5;
    const int waveM = wave & 3;        // 0..3
    const int waveN = wave >> 2;       // 0..1
    const int mBase = blockIdx.y * 128 + waveM * 32;
    const int nBase = blockIdx.x * 128 + waveN * 64;

    const int r  = lane & 15;          // row-within-fragment for A/B operands
    const int kh = (lane >> 4) << 1;   // lanes 0-15 -> K{0,1}; lanes 16-31 -> K{2,3}

    // A-fragment row pointers (2 M-tiles per wave)
    const float* aRow0 = pix + (size_t)(mBase + r) * K_TOT + kh;
    const float* aRow1 = aRow0 + (size_t)16 * K_TOT;
    // B-fragment row pointers (4 N-tiles per wave); B[k,n] = W[n,k]
    const float* bRow0 = W + (size_t)(nBase + r) * K_TOT + kh;
    const float* bRow1 = bRow0 + (size_t)16 * K_TOT;
    const float* bRow2 = bRow0 + (size_t)32 * K_TOT;
    const float* bRow3 = bRow0 + (size_t)48 * K_TOT;

    // One-time preamble: warm the full pixel-row stream in L2
    // (gfx1250 global_prefetch_b8), one prefetch per 128B line.
#pragma unroll
    for (int p = 0; p < K_TOT; p += 32) {
        __builtin_prefetch(aRow0 + p, 0, 1);
        __builtin_prefetch(aRow1 + p, 0, 1);
    }

    v8f acc[2][4];
#pragma unroll
    for (int i = 0; i < 2; ++i)
#pragma unroll
        for (int j = 0; j < 4; ++j)
            acc[i][j] = (v8f)(0.0f);

    // 3-deep software pipeline over the 147 k-steps.
    Frag fr[3];
    fr[0] = load_frag(aRow0, aRow1, bRow0, bRow1, bRow2, bRow3, 0);
    fr[1] = load_frag(aRow0, aRow1, bRow0, bRow1, bRow2, bRow3, 4);
    fr[2] = load_frag(aRow0, aRow1, bRow0, bRow1, bRow2, bRow3, 8);

    // Main loop: 144 steps (144 % 3 == 0 -> static buffer index, no guards).
#pragma unroll 3
    for (int s = 0; s < NK_STEPS - 3; ++s) {
        const int b = s % 3;
        do_mma(fr[b], acc);
        fr[b] = load_frag(aRow0, aRow1, bRow0, bRow1, bRow2, bRow3, 4 * (s + 3));
    }
    // Drain: last 3 steps, compute only.
    do_mma(fr[0], acc);
    do_mma(fr[1], acc);
    do_mma(fr[2], acc);

    // Fused epilogue: positional embedding gather + padding mask + store.
    // C/D layout: VGPR v -> row (v + 8*hi), lane%16 -> column.
    const int hi = lane >> 4;
    const float* pt0 = ptab;                        // pos_table[0]
    const float* pt1 = ptab + (size_t)64 * N_TOT;   // pos_table[1]

#pragma unroll
    for (int i = 0; i < 2; ++i) {
        const int rowBase = mBase + i * 16 + hi * 8;
#pragma unroll
        for (int v = 0; v < 8; ++v) {
            const int row = rowBase + v;
            const int2 id = ((const int2*)ids)[row];
            int i0 = id.x < 0 ? 0 : (id.x > 63 ? 63 : id.x);
            int i1 = id.y < 0 ? 0 : (id.y > 63 ? 63 : id.y);
            const bool pad = mask[row] != 0;
            const float* p0 = pt0 + (size_t)i0 * N_TOT;
            const float* p1 = pt1 + (size_t)i1 * N_TOT;
            float* orow = out + (size_t)row * N_TOT;
#pragma unroll
            for (int j = 0; j < 4; ++j) {
                const int col = nBase + j * 16 + r;
                float pos = pad ? 0.0f : (p0[col] + p1[col]);
                // output is write-once: non-temporal store keeps L2 for pix/W/ptab
                __builtin_nontemporal_store(acc[i][j][v] + pos, &orow[col]);
            }
        }
    }
}

extern "C" void kernel_launch(void* const* d_in, const int* in_sizes, int n_in,
                              void* d_out, int out_size, void* d_ws, size_t ws_size,
                              hipStream_t stream) {
    const float*         pix  = (const float*)d_in[0];         // pixel_values [B,L,PD]
    const int*           ids  = (const int*)d_in[1];           // position_ids [B,L,2]
    const unsigned char* mask = (const unsigned char*)d_in[2]; // padding_mask [B,L] (bool)
    const float*         W    = (const float*)d_in[3];         // [D,PD]
    const float*         ptab = (const float*)d_in[4];         // [2,K,D]
    float*               out  = (float*)d_out;                 // [B,L,D]

    dim3 grid(N_TOT / 128, M_TOT / 128);  // (9, 256)
    patch_embed_wmma_f32<<<grid, 256, 0, stream>>>(pix, ids, mask, W, ptab, out);
}